// MultiBoxLoss_47450798686502
// MI455X (gfx1250) — compile-verified
//
#include <hip/hip_runtime.h>

typedef __attribute__((ext_vector_type(2))) float v2f;
typedef __attribute__((ext_vector_type(8))) float v8f;

namespace mbx {
constexpr int B = 64;
constexpr int P = 24564;
constexpr int C = 21;
constexpr int O = 16;
constexpr int WB = 256;                 // priors per block in k1/k3
constexpr int NB = (P + WB - 1) / WB;   // 96
constexpr float THRESH = 0.5f;
}

using namespace mbx;

// ---------------------------------------------------------------- k0: init
__global__ void k0_init(unsigned long long* gbp, int* npos) {
  int i = blockIdx.x * blockDim.x + threadIdx.x;
  if (i < B * O) gbp[i] = 0ull;
  if (i < B)     npos[i] = 0;
}

// ------------------------------------------------- k1: overlaps + argmaxes
__global__ void k1_overlap(const float* __restrict__ priors,
                           const float* __restrict__ targets,
                           float* __restrict__ bto, int* __restrict__ bti,
                           unsigned long long* __restrict__ gbp) {
  const int b = blockIdx.y;
  const int p = blockIdx.x * WB + threadIdx.x;
  __shared__ float st[O * 5];
  __shared__ unsigned long long sbp[O];
  if (threadIdx.x < O * 5) st[threadIdx.x] = targets[(size_t)b * O * 5 + threadIdx.x];
  if (threadIdx.x < O)     sbp[threadIdx.x] = 0ull;
  __syncthreads();

  const bool valid = (p < P);
  float pxm = 0.f, pym = 0.f, pxM = 0.f, pyM = 0.f, areap = 0.f;
  if (valid) {
    float cx = priors[p * 4 + 0], cy = priors[p * 4 + 1];
    float w  = priors[p * 4 + 2], h  = priors[p * 4 + 3];
    pxm = cx - 0.5f * w; pym = cy - 0.5f * h;
    pxM = cx + 0.5f * w; pyM = cy + 0.5f * h;
    areap = w * h;
  }

  float best = -1.0f; int bestj = 0;
  float iou_arr[O];
#pragma unroll
  for (int j = 0; j < O; ++j) {
    float txm = st[j * 5 + 0], tym = st[j * 5 + 1];
    float txM = st[j * 5 + 2], tyM = st[j * 5 + 3];
    float iw = fmaxf(fminf(pxM, txM) - fmaxf(pxm, txm), 0.0f);
    float ih = fmaxf(fminf(pyM, tyM) - fmaxf(pym, tym), 0.0f);
    float inter = iw * ih;
    float at = (txM - txm) * (tyM - tym);
    float iou = valid ? (inter / (at + areap - inter)) : 0.0f;
    iou_arr[j] = iou;
    if (iou > best) { best = iou; bestj = j; }   // first-max (jnp.argmax)
  }

  if (valid) {
    bto[(size_t)b * P + p] = best;
    bti[(size_t)b * P + p] = bestj;
#pragma unroll
    for (int j = 0; j < O; ++j) {
      unsigned ib = __float_as_uint(iou_arr[j]);          // iou >= 0 -> monotonic bits
      unsigned long long pk =
          ((unsigned long long)ib << 32) |
          (unsigned long long)(0xFFFFFFFFu - (unsigned)p); // tie-break: smallest p wins
      atomicMax(&sbp[j], pk);
    }
  }
  __syncthreads();
  if (threadIdx.x < O) atomicMax(&gbp[b * O + threadIdx.x], sbp[threadIdx.x]);
}

// ------------------------------------ k2: sequential best-prior overrides
__global__ void k2_override(const unsigned long long* __restrict__ gbp,
                            float* __restrict__ bto, int* __restrict__ bti) {
  int b = blockIdx.x * blockDim.x + threadIdx.x;
  if (b >= B) return;
  for (int j = 0; j < O; ++j) {                 // in order: later j wins
    unsigned long long pk = gbp[b * O + j];
    unsigned p = 0xFFFFFFFFu - (unsigned)(pk & 0xFFFFFFFFull);
    bto[(size_t)b * P + p] = 2.0f;
    bti[(size_t)b * P + p] = j;
  }
}

// ------------------------- k3: main streaming pass (WMMA sum-exp per 16 priors)
__global__ void k3_main(const float* __restrict__ loc,
                        const float* __restrict__ conf,
                        const float* __restrict__ priors,
                        const float* __restrict__ targets,
                        const float* __restrict__ bto,
                        const int* __restrict__ bti,
                        float* __restrict__ lmine,
                        float* __restrict__ part_ll,
                        float* __restrict__ part_lcp,
                        int* __restrict__ npos) {
  const int b    = blockIdx.y;
  const int base = blockIdx.x * WB;
  const int tid  = threadIdx.x;
  const int lane = tid & 31;
  const int wave = tid >> 5;
  const int r    = lane & 15;   // prior row within 16-group
  const int half = lane >> 4;   // 0 -> K classes {0,1} of each chunk; 1 -> {2,3}

  float ll_acc = 0.0f, lcp_acc = 0.0f;
  int   np_acc = 0;

  for (int g = 0; g < 2; ++g) {
    const int gp = base + (wave * 2 + g) * 16;
    const int p  = gp + r;
    const bool valid = (p < P);
    const size_t row = ((size_t)b * P + (size_t)(valid ? p : 0)) * C;

    // prefetch next group's conf rows (speculative, safe OOB)
    __builtin_prefetch(conf + row + (size_t)16 * C, 0, 0);

    // Each lane loads 12 of the (padded-to-24) class logits for its prior.
    float v[12];
#pragma unroll
    for (int kc = 0; kc < 6; ++kc) {
#pragma unroll
      for (int t = 0; t < 2; ++t) {
        const int c = 4 * kc + 2 * half + t;
        float x;
        if (c < C) x = valid ? conf[row + c] : 0.0f;
        else       x = -1e30f;                      // exp -> 0
        v[2 * kc + t] = x;
      }
    }

    // Row max across the lane pair (lane L <-> L^16 hold the same prior).
    float mymax = v[0];
#pragma unroll
    for (int i = 1; i < 12; ++i) mymax = fmaxf(mymax, v[i]);
    float mx = fmaxf(mymax, __shfl_xor(mymax, 16, 32));

    // Sum of exp over 21 classes for 16 priors via 6 chained
    // v_wmma_f32_16x16x4_f32 against an all-ones B matrix:
    // D[m][n] = sum_k A[m][k].  A layout: lanes 0-15 K={0,1}, lanes 16-31 K={2,3}.
    v8f acc = {0.f, 0.f, 0.f, 0.f, 0.f, 0.f, 0.f, 0.f};
    const v2f ones = {1.0f, 1.0f};
#pragma unroll
    for (int kc = 0; kc < 6; ++kc) {
      v2f a;
      a.x = expf(v[2 * kc + 0] - mx);
      a.y = expf(v[2 * kc + 1] - mx);
      acc = __builtin_amdgcn_wmma_f32_16x16x4_f32(
          /*neg_a=*/false, a, /*neg_b=*/false, ones,
          /*c_mod=*/(short)0, acc, /*reuse_a=*/false, /*reuse_b=*/false);
    }

    // Recover rowsum(prior r): lanes 0-15 hold rows 0-7 in VGPRs 0-7,
    // lanes 16-31 hold rows 8-15.  One swap with the partner lane covers all.
    float s_own = acc[r & 7];
    float s_sw  = __shfl_xor(s_own, 16, 32);
    bool  own_ok = (half == ((r >> 3) & 1));
    float sumexp = own_ok ? s_own : s_sw;
    float lse = mx + logf(sumexp);

    if (valid && half == 0) {
      const size_t ip = (size_t)b * P + p;
      float o  = bto[ip];
      int   ti = bti[ip];
      const float* tb = targets + ((size_t)b * O + ti) * 5;
      float label = tb[4];
      int   ct  = (o < THRESH) ? 0 : ((int)label + 1);
      bool  pos = ct > 0;

      if (pos) {
        float pcx = priors[p * 4 + 0], pcy = priors[p * 4 + 1];
        float pw  = priors[p * 4 + 2], ph  = priors[p * 4 + 3];
        float xm = tb[0], ym = tb[1], xM = tb[2], yM = tb[3];
        float g4[4];
        g4[0] = ((xm + xM) * 0.5f - pcx) / (0.1f * pw);
        g4[1] = ((ym + yM) * 0.5f - pcy) / (0.1f * ph);
        g4[2] = logf((xM - xm) / pw) * 5.0f;   // /0.2
        g4[3] = logf((yM - ym) / ph) * 5.0f;
        const float* lp = loc + ip * 4;
#pragma unroll
        for (int q = 0; q < 4; ++q) {
          float d = lp[q] - g4[q];
          float ad = fabsf(d);
          ll_acc += (ad < 1.0f) ? (0.5f * d * d) : (ad - 0.5f);
        }
      }
      float gathered = conf[row + ct];
      float lca = lse - gathered;               // >= 0 always
      lmine[ip] = pos ? 0.0f : lca;
      if (pos) { lcp_acc += lca; np_acc += 1; }
    }
  }

  // Deterministic block tree reductions.
  __shared__ float sred[WB];
  __shared__ int   sredi[WB];
  const int pidx = b * gridDim.x + blockIdx.x;

  sred[tid] = ll_acc; __syncthreads();
  for (int s = WB / 2; s > 0; s >>= 1) { if (tid < s) sred[tid] += sred[tid + s]; __syncthreads(); }
  if (tid == 0) part_ll[pidx] = sred[0];
  __syncthreads();

  sred[tid] = lcp_acc; __syncthreads();
  for (int s = WB / 2; s > 0; s >>= 1) { if (tid < s) sred[tid] += sred[tid + s]; __syncthreads(); }
  if (tid == 0) part_lcp[pidx] = sred[0];
  __syncthreads();

  sredi[tid] = np_acc; __syncthreads();
  for (int s = WB / 2; s > 0; s >>= 1) { if (tid < s) sredi[tid] += sredi[tid + s]; __syncthreads(); }
  if (tid == 0) atomicAdd(&npos[b], sredi[0]);   // int: deterministic
}

// ----------------------- k4: exact per-batch top-k SUM via radix-select
__global__ void k4_topk(const float* __restrict__ lmine,
                        const int* __restrict__ npos,
                        float* __restrict__ topk) {
  const int b = blockIdx.x;
  const float* lm = lmine + (size_t)b * P;
  const int tid = threadIdx.x;
  __shared__ int   h[16];
  __shared__ float sred[1024];

  long long k64 = 3LL * (long long)npos[b];
  if (k64 > P - 1) k64 = P - 1;
  if (k64 <= 0) { if (tid == 0) topk[b] = 0.0f; return; }   // uniform branch
  int kk = (int)k64;

  unsigned prefix = 0;
  for (int pass = 0; pass < 8; ++pass) {
    const int shift = 28 - pass * 4;
    const unsigned hi = (pass == 0) ? 0u : (0xFFFFFFFFu << (shift + 4));
    if (tid < 16) h[tid] = 0;
    __syncthreads();
    for (int i = tid; i < P; i += 1024) {
      unsigned bits = __float_as_uint(lm[i]);   // lm >= 0 -> bits monotonic
      if ((bits & hi) == prefix) atomicAdd(&h[(bits >> shift) & 15], 1);
    }
    __syncthreads();
    int rem = kk;
    int d = 15;
    for (; d > 0; --d) { if (rem > h[d]) rem -= h[d]; else break; }
    prefix |= (unsigned)d << shift;
    kk = rem;
    __syncthreads();
  }

  const float T = __uint_as_float(prefix);      // k-th largest value
  float part = 0.0f;
  for (int i = tid; i < P; i += 1024) {
    float v = lm[i];
    if (__float_as_uint(v) > prefix) part += v;
  }
  sred[tid] = part; __syncthreads();
  for (int s = 512; s > 0; s >>= 1) { if (tid < s) sred[tid] += sred[tid + s]; __syncthreads(); }
  if (tid == 0) topk[b] = sred[0] + (float)kk * T;
}

// ------------------------------------------------ k5: final reductions
__global__ void k5_final(const float* __restrict__ part_ll,
                         const float* __restrict__ part_lcp,
                         int nparts,
                         const float* __restrict__ topk,
                         const int* __restrict__ npos,
                         float* __restrict__ out) {
  __shared__ float s[256];
  __shared__ int  si[256];
  const int tid = threadIdx.x;

  float a = 0.0f;
  for (int i = tid; i < nparts; i += 256) a += part_ll[i];
  s[tid] = a; __syncthreads();
  for (int st = 128; st > 0; st >>= 1) { if (tid < st) s[tid] += s[tid + st]; __syncthreads(); }
  float loss_l = s[0];
  __syncthreads();

  float c = 0.0f;
  for (int i = tid; i < nparts; i += 256) c += part_lcp[i];
  if (tid < B) c += topk[tid];
  s[tid] = c; __syncthreads();
  for (int st = 128; st > 0; st >>= 1) { if (tid < st) s[tid] += s[tid + st]; __syncthreads(); }
  float loss_c = s[0];
  __syncthreads();

  int n = 0;
  for (int i = tid; i < B; i += 256) n += npos[i];
  si[tid] = n; __syncthreads();
  for (int st = 128; st > 0; st >>= 1) { if (tid < st) si[tid] += si[tid + st]; __syncthreads(); }

  if (tid == 0) {
    float N = (float)si[0];
    out[0] = loss_l / N;
    out[1] = loss_c / N;
  }
}

// -------------------------------------------------------------- launcher
extern "C" void kernel_launch(void* const* d_in, const int* in_sizes, int n_in,
                              void* d_out, int out_size, void* d_ws, size_t ws_size,
                              hipStream_t stream) {
  (void)in_sizes; (void)n_in; (void)out_size; (void)ws_size;
  const float* loc     = (const float*)d_in[0];
  const float* conf    = (const float*)d_in[1];
  const float* priors  = (const float*)d_in[2];
  const float* targets = (const float*)d_in[3];
  float* out = (float*)d_out;

  char* ws = (char*)d_ws;
  size_t off = 0;
  auto carve = [&](size_t bytes) -> void* {
    void* p = ws + off;
    off += (bytes + 255) & ~(size_t)255;
    return p;
  };
  float*              bto      = (float*)carve((size_t)B * P * sizeof(float));
  int*                bti      = (int*)  carve((size_t)B * P * sizeof(int));
  float*              lmine    = (float*)carve((size_t)B * P * sizeof(float));
  unsigned long long* gbp      = (unsigned long long*)carve((size_t)B * O * 8);
  int*                npos     = (int*)  carve((size_t)B * sizeof(int));
  float*              part_ll  = (float*)carve((size_t)B * NB * sizeof(float));
  float*              part_lcp = (float*)carve((size_t)B * NB * sizeof(float));
  float*              topk     = (float*)carve((size_t)B * sizeof(float));

  k0_init<<<(B * O + 255) / 256, 256, 0, stream>>>(gbp, npos);
  k1_overlap<<<dim3(NB, B), WB, 0, stream>>>(priors, targets, bto, bti, gbp);
  k2_override<<<1, 64, 0, stream>>>(gbp, bto, bti);
  k3_main<<<dim3(NB, B), WB, 0, stream>>>(loc, conf, priors, targets,
                                          bto, bti, lmine, part_ll, part_lcp, npos);
  k4_topk<<<B, 1024, 0, stream>>>(lmine, npos, topk);
  k5_final<<<1, 256, 0, stream>>>(part_ll, part_lcp, B * NB, topk, npos, out);
}